// Word2Vec_76304388981348
// MI455X (gfx1250) — compile-verified
//
#include <hip/hip_runtime.h>
#include <hip/hip_bf16.h>

typedef __attribute__((ext_vector_type(2))) float v2f;
typedef __attribute__((ext_vector_type(8))) float v8f;

#define EMB  128
#define NEGS 10

__device__ __forceinline__ float sel8(v8f v, int j) {
  float r = v[0];
  r = (j == 1) ? v[1] : r;
  r = (j == 2) ? v[2] : r;
  r = (j == 3) ? v[3] : r;
  r = (j == 4) ? v[4] : r;
  r = (j == 5) ? v[5] : r;
  r = (j == 6) ? v[6] : r;
  r = (j == 7) ? v[7] : r;
  return r;
}

__device__ __forceinline__ float log_sigmoid(float x) {
  // stable: min(x,0) - log1p(exp(-|x|))
  return fminf(x, 0.0f) - log1pf(expf(-fabsf(x)));
}

// One wave (32 lanes) handles a tile of 16 batch elements.
// A-fragment (16x4 f32) and B-fragment (4x16 f32) share the same lane mapping:
//   row  = lane & 15
//   koff = 4*kc + 2*(lane>>4) .. +1
// so each lane loads float2's straight from the gathered embedding rows.
__global__ void __launch_bounds__(256, 2) w2v_loss_kernel(
    const float* __restrict__ iemb, const float* __restrict__ oemb,
    const int* __restrict__ iwords, const int* __restrict__ owords,
    const int* __restrict__ negwords, float* __restrict__ partial)
{
  const int lane  = threadIdx.x & 31;
  const int wave  = threadIdx.x >> 5;
  const int base  = (blockIdx.x * 8 + wave) * 16;   // 16 batch rows per wave
  const int r     = lane & 15;
  const int b     = base + r;                        // batch element this lane serves
  const int khalf = (lane >> 4) << 1;                // 0 or 2

  const float* irow = iemb + (size_t)iwords[b] * EMB;
  const float* orow = oemb + (size_t)owords[b] * EMB;
  const float* nrow[NEGS];
#pragma unroll
  for (int j = 0; j < NEGS; ++j)
    nrow[j] = oemb + (size_t)negwords[b * NEGS + j] * EMB;

  v8f accp = {0.f, 0.f, 0.f, 0.f, 0.f, 0.f, 0.f, 0.f};
  v8f accn = {0.f, 0.f, 0.f, 0.f, 0.f, 0.f, 0.f, 0.f};

#pragma unroll 2
  for (int kc = 0; kc < EMB / 4; ++kc) {
    const int ko = kc * 4 + khalf;
    v2f a  = *(const v2f*)(irow + ko);   // A: iemb tile
    v2f bp = *(const v2f*)(orow + ko);   // B: oemb tile (transposed by layout)
    v2f bn = *(const v2f*)(nrow[0] + ko);
#pragma unroll
    for (int j = 1; j < NEGS; ++j) {     // sum of 10 neg rows (k-sum commuted)
      v2f t = *(const v2f*)(nrow[j] + ko);
      bn.x += t.x;
      bn.y += t.y;
    }
    accp = __builtin_amdgcn_wmma_f32_16x16x4_f32(false, a, false, bp,
                                                 (short)0, accp, false, false);
    accn = __builtin_amdgcn_wmma_f32_16x16x4_f32(false, a, false, bn,
                                                 (short)0, accn, false, false);
  }

  // Diagonal of D: lanes 0..7 -> m=lane (vgpr lane), lanes 24..31 -> m=lane-16 (vgpr lane-24)
  const bool valid = (lane < 8) || (lane >= 24);
  const int  j     = (lane < 8) ? lane : ((lane >= 24) ? (lane - 24) : 0);
  const float ps = sel8(accp, j);   // dot(iemb[m], oemb[m])
  const float ns = sel8(accn, j);   // sum_k dot(iemb[m], neg_o[m,k])
  float contrib = valid ? -(log_sigmoid(ps) + log_sigmoid(-ns)) : 0.0f;

#pragma unroll
  for (int off = 16; off > 0; off >>= 1)
    contrib += __shfl_down(contrib, off, 32);

  __shared__ float sred[8];
  if (lane == 0) sred[wave] = contrib;
  __syncthreads();
  if (threadIdx.x == 0) {
    float s = 0.f;
#pragma unroll
    for (int w = 0; w < 8; ++w) s += sred[w];
    partial[blockIdx.x] = s;
  }
}

// Deterministic final reduction over per-block partials (n <= 128).
__global__ void w2v_reduce_kernel(const float* __restrict__ partial,
                                  float* __restrict__ out, int n) {
  float v = ((int)threadIdx.x < n) ? partial[threadIdx.x] : 0.0f;
#pragma unroll
  for (int off = 16; off > 0; off >>= 1)
    v += __shfl_down(v, off, 32);
  __shared__ float s[4];
  const int lane = threadIdx.x & 31;
  const int w    = threadIdx.x >> 5;
  if (lane == 0) s[w] = v;
  __syncthreads();
  if (threadIdx.x == 0) out[0] = s[0] + s[1] + s[2] + s[3];
}

extern "C" void kernel_launch(void* const* d_in, const int* in_sizes, int n_in,
                              void* d_out, int out_size, void* d_ws, size_t ws_size,
                              hipStream_t stream) {
  const float* iemb = (const float*)d_in[0];   // [VOCAB, 128] f32
  const float* oemb = (const float*)d_in[1];   // [VOCAB, 128] f32
  const int*   iw   = (const int*)d_in[2];     // [B]
  const int*   ow   = (const int*)d_in[3];     // [B]
  const int*   nw   = (const int*)d_in[4];     // [B, 10]
  float* out     = (float*)d_out;              // scalar loss
  float* partial = (float*)d_ws;               // one float per block

  const int B      = in_sizes[2];              // 16384
  const int blocks = B / 128;                  // 8 waves * 16 rows per block

  w2v_loss_kernel<<<blocks, 256, 0, stream>>>(iemb, oemb, iw, ow, nw, partial);
  w2v_reduce_kernel<<<1, 128, 0, stream>>>(partial, out, blocks);
}